// AdaIN_52321291600115
// MI455X (gfx1250) — compile-verified
//
#include <hip/hip_runtime.h>

// ---------------------------------------------------------------------------
// AdaIN (semantic, per-(batch,class)) for MI455X / gfx1250.
//
// Phase 1 (stats_kernel):   segmented count/sum/sumsq via V_WMMA_F32_16X16X4_F32
//                           (one-hot A x feature B, exact fp32), per-wave VGPR
//                           accumulators, flushed with global float atomics.
//                           High-class tile (classes 16..18 padded to 16..31)
//                           is skipped via a wave-uniform branch when no label
//                           >= 16 appears in the 4-pixel group (~50% of iters),
//                           keeping EXEC all-ones around every WMMA.
// Phase 2 (finalize_kernel):mean / unbiased std / valid; folds the affine into
//                           per-(b,k,c) (scale,shift); writes secondary outputs.
// Phase 3 (apply_kernel):   out = x*scale + shift, float4-vectorized.
// ---------------------------------------------------------------------------

typedef __attribute__((ext_vector_type(2))) float v2f;
typedef __attribute__((ext_vector_type(8))) float v8f;

constexpr int BB   = 4;
constexpr int CC   = 64;
constexpr int KK   = 19;          // real classes
constexpr int KP   = 32;          // padded classes (2 WMMA M-tiles)
constexpr int HW   = 512 * 512;   // pixels per image

// Workspace float offsets
constexpr int WS_CNT = 0;                      // [B*K]        counts
constexpr int WS_S1  = WS_CNT + BB * KK;       // [B*KP*C]     sums
constexpr int WS_S2  = WS_S1 + BB * KP * CC;   // [B*KP*C]     sums of squares
constexpr int WS_SS  = WS_S2 + BB * KP * CC;   // [B*K*C] float2 (scale,shift)
constexpr int WS_ZERO_N = WS_SS;               // floats to zero each launch

__global__ __launch_bounds__(256) void zero_kernel(float* __restrict__ ws, int n) {
    int i = blockIdx.x * blockDim.x + threadIdx.x;
    if (i < n) ws[i] = 0.0f;
}

// ---------------------------------------------------------------------------
// Stats: grid = 256 blocks x 256 threads = 2048 waves; 512 waves per batch;
// each wave owns 512 consecutive pixels of one image and runs 128 iterations
// of 4 pixels: 8 unconditional WMMAs (classes 0..15, S1+S2 over 4 channel
// tiles) plus 8 more only when a label >= 16 is present (uniform branch).
//
// WMMA f32 16x16x4 operand mapping (wave32):
//   A (16x4):  lanes 0-15 hold M=lane: VGPR0=K0, VGPR1=K1; lanes 16-31: K2,K3
//   B (4x16):  lanes 0-15 hold N=lane: VGPR0=K0, VGPR1=K1; lanes 16-31: K2,K3
//   D (16x16): VGPR j: lanes0-15 -> (M=j, N=lane); lanes16-31 -> (M=j+8)
// ---------------------------------------------------------------------------
__global__ __launch_bounds__(256) void stats_kernel(const float* __restrict__ x,
                                                    const int*   __restrict__ y,
                                                    float*       __restrict__ ws) {
    const int tid  = threadIdx.x;
    const int lane = tid & 31;
    const int warp = tid >> 5;
    const int half = lane >> 4;   // 0: K=0,1 slots; 1: K=2,3 slots
    const int q    = lane & 15;   // M (class row) for A, N (channel col) for B

    const int b    = blockIdx.x >> 6;                 // 64 blocks per batch
    const int wave = ((blockIdx.x & 63) << 3) + warp; // wave id within batch
    const int p0   = wave * 512;

    const float* xb = x + (size_t)b * CC * HW;
    const int*   yb = y + (size_t)b * HW;

    v8f s1[2][4] = {};   // [class tile][channel tile] accumulators
    v8f s2[2][4] = {};
    float c0 = 0.0f, c1 = 0.0f;   // per-lane class counts (classes q, q+16)

    for (int p = p0; p < p0 + 512; p += 4) {
        // Labels for the two pixels this half-lane feeds (K slots 2h, 2h+1).
        const int2 lab2 = *(const int2*)(yb + p + 2 * half);
        const int la = lab2.x;
        const int lb = lab2.y;

        v2f A0;   // one-hot tile for classes 0..15 (row = q)
        A0.x = (la == q) ? 1.0f : 0.0f;
        A0.y = (lb == q) ? 1.0f : 0.0f;
        c0 += A0.x + A0.y;

        v2f Bsave[4], Bqsave[4];
#pragma unroll
        for (int g = 0; g < 4; ++g) {           // channel tiles of 16
            const float* xc = xb + (size_t)(16 * g + q) * HW;
            v2f Bv, Bq;
            Bv.x = xc[p + 2 * half];
            Bv.y = xc[p + 2 * half + 1];
            Bq.x = Bv.x * Bv.x;
            Bq.y = Bv.y * Bv.y;
            Bsave[g]  = Bv;
            Bqsave[g] = Bq;

            s1[0][g] = __builtin_amdgcn_wmma_f32_16x16x4_f32(
                false, A0, false, Bv, (short)0, s1[0][g], false, false);
            s2[0][g] = __builtin_amdgcn_wmma_f32_16x16x4_f32(
                false, A0, false, Bq, (short)0, s2[0][g], false, false);
        }

        // High-class tile: only ~50% of 4-pixel groups contain a label >= 16.
        // __any() is uniform across the wave; readfirstlane makes it scalar so
        // the branch is an s_cbranch and EXEC stays all-ones for the WMMAs.
        const int hot = __any((la >= 16) || (lb >= 16));
        if (__builtin_amdgcn_readfirstlane(hot)) {
            v2f A1;   // one-hot tile for classes 16..31 (row = q+16)
            A1.x = (la == q + 16) ? 1.0f : 0.0f;
            A1.y = (lb == q + 16) ? 1.0f : 0.0f;
            c1 += A1.x + A1.y;
#pragma unroll
            for (int g = 0; g < 4; ++g) {
                s1[1][g] = __builtin_amdgcn_wmma_f32_16x16x4_f32(
                    false, A1, false, Bsave[g], (short)0, s1[1][g], false, false);
                s2[1][g] = __builtin_amdgcn_wmma_f32_16x16x4_f32(
                    false, A1, false, Bqsave[g], (short)0, s2[1][g], false, false);
            }
        }
    }

    // ---- flush (divergence is fine now; all WMMAs done) ----
    float* cnt = ws + WS_CNT;
    float* S1  = ws + WS_S1;
    float* S2  = ws + WS_S2;

    atomicAdd(&cnt[b * KK + q], c0);              // classes 0..15 (2 lanes each)
    if (q < KK - 16)                              // classes 16..18
        atomicAdd(&cnt[b * KK + 16 + q], c1);

#pragma unroll
    for (int t = 0; t < 2; ++t)
#pragma unroll
        for (int g = 0; g < 4; ++g)
#pragma unroll
            for (int j = 0; j < 8; ++j) {
                const int m   = 16 * t + j + 8 * half;   // padded class 0..31
                const int n   = 16 * g + q;              // channel 0..63
                const int idx = (b * KP + m) * CC + n;
                atomicAdd(&S1[idx], s1[t][g][j]);
                atomicAdd(&S2[idx], s2[t][g][j]);
            }
}

// ---------------------------------------------------------------------------
// Finalize: one thread per (b,k,c). Folds (x-mean)/std*sstd+smean into a
// single fma: out = x*scale + shift  (scale=1, shift=0 where invalid).
// ---------------------------------------------------------------------------
__global__ __launch_bounds__(256) void finalize_kernel(const float* __restrict__ sm,
                                                       const float* __restrict__ ssd,
                                                       float* __restrict__ ws,
                                                       float* __restrict__ out2,
                                                       float* __restrict__ out3,
                                                       float* __restrict__ out4) {
    const int idx = blockIdx.x * blockDim.x + threadIdx.x;
    if (idx >= BB * KK * CC) return;
    const int c  = idx % CC;
    const int bk = idx / CC;
    const int k  = bk % KK;
    const int b  = bk / KK;

    const float cntv = ws[WS_CNT + b * KK + k];
    const float s1v  = ws[WS_S1 + (b * KP + k) * CC + c];
    const float s2v  = ws[WS_S2 + (b * KP + k) * CC + c];

    const float cs   = fmaxf(cntv, 1.0f);
    const float mean = s1v / cs;
    const float var  = (s2v - cs * mean * mean) / fmaxf(cntv - 1.0f, 1.0f);
    const float sd   = sqrtf(fmaxf(var, 0.0f)) + 1e-5f;
    const bool valid = cntv > 6.0f;

    const float stm = sm[k * CC + c];
    const float sts = ssd[k * CC + c];

    const float scale = valid ? (sts / sd) : 1.0f;
    const float shift = valid ? (stm - mean * scale) : 0.0f;

    float2* ssbuf = (float2*)(ws + WS_SS);
    ssbuf[(b * KK + k) * CC + c] = make_float2(scale, shift);

    out2[idx] = valid ? stm : 0.0f;      // style_means_1dim
    out3[idx] = valid ? sts : 0.0f;      // style_stds_1dim
    if (c == 0) out4[b * KK + k] = valid ? 1.0f : 0.0f;   // valid_bk
}

// ---------------------------------------------------------------------------
// Apply: grid = 1024 blocks x 256 threads. Block (b, strip) owns pixels
// [strip*1024, strip*1024+1024) for all 64 channels. Each thread holds 4
// pixels; labels loaded once, then 64 channel iterations of float4 in/out.
// ---------------------------------------------------------------------------
__global__ __launch_bounds__(256) void apply_kernel(const float* __restrict__ x,
                                                    const int*   __restrict__ y,
                                                    const float* __restrict__ ws,
                                                    float*       __restrict__ out) {
    const int tid   = threadIdx.x;
    const int b     = blockIdx.x >> 8;      // 256 strips per batch
    const int strip = blockIdx.x & 255;
    const int p     = strip * 1024 + tid * 4;

    const int4 l4 = *(const int4*)(y + (size_t)b * HW + p);
    const int o0 = (b * KK + l4.x) * CC;
    const int o1 = (b * KK + l4.y) * CC;
    const int o2 = (b * KK + l4.z) * CC;
    const int o3 = (b * KK + l4.w) * CC;

    const float2* ssbuf = (const float2*)(ws + WS_SS);

    for (int c = 0; c < CC; ++c) {
        const size_t base = ((size_t)(b * CC + c)) * HW + p;
        const float4 v = *(const float4*)(x + base);
        const float2 t0 = ssbuf[o0 + c];
        const float2 t1 = ssbuf[o1 + c];
        const float2 t2 = ssbuf[o2 + c];
        const float2 t3 = ssbuf[o3 + c];
        float4 r;
        r.x = fmaf(v.x, t0.x, t0.y);
        r.y = fmaf(v.y, t1.x, t1.y);
        r.z = fmaf(v.z, t2.x, t2.y);
        r.w = fmaf(v.w, t3.x, t3.y);
        *(float4*)(out + base) = r;
    }
}

extern "C" void kernel_launch(void* const* d_in, const int* in_sizes, int n_in,
                              void* d_out, int out_size, void* d_ws, size_t ws_size,
                              hipStream_t stream) {
    const float* x  = (const float*)d_in[0];   // x_content  [B,C,H,W] f32
    const int*   y  = (const int*)  d_in[1];   // y_content  [B,H,W]   i32
    const float* sm = (const float*)d_in[2];   // style_means [K,C]    f32
    const float* ss = (const float*)d_in[3];   // style_stds  [K,C]    f32

    float* ws  = (float*)d_ws;
    float* out = (float*)d_out;

    const int O1 = BB * CC * HW;               // main output elements
    float* out2 = out + O1;                    // style_means_1dim [B,K,C]
    float* out3 = out2 + BB * KK * CC;         // style_stds_1dim  [B,K,C]
    float* out4 = out3 + BB * KK * CC;         // valid_bk         [B,K]

    zero_kernel<<<(WS_ZERO_N + 255) / 256, 256, 0, stream>>>(ws, WS_ZERO_N);
    stats_kernel<<<256, 256, 0, stream>>>(x, y, ws);
    finalize_kernel<<<(BB * KK * CC + 255) / 256, 256, 0, stream>>>(sm, ss, ws,
                                                                    out2, out3, out4);
    apply_kernel<<<1024, 256, 0, stream>>>(x, y, ws, out);
}